// SymmetryAwareRotationalLoss_51926154609200
// MI455X (gfx1250) — compile-verified
//
#include <hip/hip_runtime.h>
#include <math.h>

// CDNA5 / gfx1250: wave32, WMMA F32_16X16X4_F32 used for the N×4 · 4×24
// symmetry einsum. Everything else is a bandwidth-bound reduction.

typedef float v2f __attribute__((ext_vector_type(2)));
typedef float v8f __attribute__((ext_vector_type(8)));

#ifndef __has_builtin
#define __has_builtin(x) 0
#endif
#define HAVE_WMMA_F32_X4 __has_builtin(__builtin_amdgcn_wmma_f32_16x16x4_f32)

namespace {
constexpr int   kW     = 256;
constexpr int   kH     = 256;
constexpr int   kHW    = kW * kH;            // 65536
constexpr int   kNpix  = 16 * kHW;           // 1,048,576 pixels (B=16)
constexpr int   kSyms  = 24;
constexpr float kEps   = 1e-6f;
constexpr float kWGrad = 0.05f;
}

__global__ __launch_bounds__(256, 2)
void rotloss_partials(const float* __restrict__ qp,
                      const float* __restrict__ qt,
                      const float* __restrict__ syms,
                      float* __restrict__ partials)
{
    const int n    = blockIdx.x * 256 + threadIdx.x;  // flat pixel id (b,h,w)
    const int lane = threadIdx.x & 31;
    const int wid  = threadIdx.x >> 5;

    const int b    = n >> 16;          // HW = 2^16
    const int pos  = n & (kHW - 1);
    const int h    = pos >> 8;         // W = 2^8
    const int w    = pos & (kW - 1);
    const int base = b * 4 * kHW;

    // ---------------- rotation loss: load + normalize + relative quat -----
    float p[4], t[4];
#pragma unroll
    for (int c = 0; c < 4; ++c) {
        p[c] = qp[base + c * kHW + pos];
        t[c] = qt[base + c * kHW + pos];
    }
    const float np = sqrtf(p[0]*p[0] + p[1]*p[1] + p[2]*p[2] + p[3]*p[3]);
    const float nt = sqrtf(t[0]*t[0] + t[1]*t[1] + t[2]*t[2] + t[3]*t[3]);
    const float ip = 1.0f / fmaxf(np, kEps);
    const float it = 1.0f / fmaxf(nt, kEps);
#pragma unroll
    for (int c = 0; c < 4; ++c) { p[c] *= ip; t[c] *= it; }

    const float w1 = p[0], x1 = p[1], y1 = p[2], z1 = p[3];
    const float w2 = t[0], x2 = t[1], y2 = t[2], z2 = t[3];
    const float rw =  w2*w1 + x2*x1 + y2*y1 + z2*z1;
    const float rx = -w2*x1 + x2*w1 - y2*z1 + z2*y1;
    const float ry = -w2*y1 + x2*z1 + y2*w1 - z2*x1;
    const float rz = -w2*z1 - x2*y1 + y2*x1 + z2*w1;

    float rot_sum = 0.0f;

#if HAVE_WMMA_F32_X4
    // A-layout for V_WMMA_F32_16X16X4_F32 (32-bit A, 16x4):
    //   lanes 0-15 : M=lane,    VGPR0=K0, VGPR1=K1
    //   lanes 16-31: M=lane-16, VGPR0=K2, VGPR1=K3
    // Half-swap so both lane groups can source both pixel groups.
    const float rw_s = __shfl_xor(rw, 16, 32);
    const float rx_s = __shfl_xor(rx, 16, 32);
    const float ry_s = __shfl_xor(ry, 16, 32);
    const float rz_s = __shfl_xor(rz, 16, 32);
    const bool  lo   = lane < 16;

    v2f A0, A1;
    A0.x = lo ? rw   : ry_s;   // tile0: rows = pixels [p .. p+15]
    A0.y = lo ? rx   : rz_s;
    A1.x = lo ? rw_s : ry;     // tile1: rows = pixels [p+16 .. p+31]
    A1.y = lo ? rx_s : rz;

    // B-layout (4x16, f32): lane%16 = column (sym index), halves/VGPRs = K,
    // with the (1,-1,-1,-1) sign fold applied. 24 syms -> two column tiles.
    const int   col  = lane & 15;
    const int   kc   = lo ? 0 : 2;
    const float sgn0 = lo ? 1.0f : -1.0f;  // signs[0]=+1, signs[1..3]=-1
    v2f B0, B1;
    B0.x = syms[col * 4 + kc]     * sgn0;
    B0.y = syms[col * 4 + kc + 1] * -1.0f;
    const int s1 = col + 16;
    if (s1 < kSyms) {
        B1.x = syms[s1 * 4 + kc]     * sgn0;
        B1.y = syms[s1 * 4 + kc + 1] * -1.0f;
    } else {
        B1.x = 0.0f; B1.y = 0.0f;          // zero-padded columns: |0| is a no-op for max
    }

    const v8f zero = {};
    v8f D00 = __builtin_amdgcn_wmma_f32_16x16x4_f32(false, A0, false, B0, (short)0, zero, false, false);
    v8f D01 = __builtin_amdgcn_wmma_f32_16x16x4_f32(false, A0, false, B1, (short)0, zero, false, false);
    v8f D10 = __builtin_amdgcn_wmma_f32_16x16x4_f32(false, A1, false, B0, (short)0, zero, false, false);
    v8f D11 = __builtin_amdgcn_wmma_f32_16x16x4_f32(false, A1, false, B1, (short)0, zero, false, false);

    // D-layout: VGPR v, lanes 0-15 -> row v, lanes 16-31 -> row v+8; col = lane%16.
    // Max over sym columns = butterfly max within each 16-lane half.
    float m0[8], m1[8];
#pragma unroll
    for (int v = 0; v < 8; ++v) {
        m0[v] = fmaxf(fabsf(D00[v]), fabsf(D01[v]));
        m1[v] = fmaxf(fabsf(D10[v]), fabsf(D11[v]));
    }
#pragma unroll
    for (int off = 1; off <= 8; off <<= 1) {
#pragma unroll
        for (int v = 0; v < 8; ++v) {
            m0[v] = fmaxf(m0[v], __shfl_xor(m0[v], off, 32));
            m1[v] = fmaxf(m1[v], __shfl_xor(m1[v], off, 32));
        }
    }
#pragma unroll
    for (int v = 0; v < 8; ++v) {
        rot_sum += 2.0f * acosf(fminf(m0[v], 1.0f - kEps));
        rot_sum += 2.0f * acosf(fminf(m1[v], 1.0f - kEps));
    }
    // lanes 0-15 hold pixels {0..7,16..23}; lanes 16-31 hold {8..15,24..31};
    // one cross-half add makes rot_sum the wave total (uniform in all lanes).
    rot_sum += __shfl_xor(rot_sum, 16, 32);
#else
    // VALU fallback (no f32 x4 WMMA builtin on this toolchain)
    float mx = 0.0f;
#pragma unroll
    for (int s = 0; s < kSyms; ++s) {
        const float d = rw * syms[s*4+0] - rx * syms[s*4+1]
                      - ry * syms[s*4+2] - rz * syms[s*4+3];
        mx = fmaxf(mx, fabsf(d));
    }
    rot_sum = 2.0f * acosf(fminf(mx, 1.0f - kEps));
#pragma unroll
    for (int off = 16; off >= 1; off >>= 1) rot_sum += __shfl_xor(rot_sum, off, 32);
#endif

    // ---------------- gradient loss (edge-padded finite differences) ------
    const int we = (w < kW - 1) ? w : kW - 2;
    const int he = (h < kH - 1) ? h : kH - 2;
    float grad_sum = 0.0f;
#pragma unroll
    for (int c = 0; c < 4; ++c) {
        const int cb = base + c * kHW;
        const float pa = qp[cb + h  * kW + we];
        const float pb = qp[cb + h  * kW + we + 1];
        const float pc = qp[cb + he * kW + w];
        const float pd = qp[cb + (he + 1) * kW + w];
        const float ta = qt[cb + h  * kW + we];
        const float tb = qt[cb + h  * kW + we + 1];
        const float tc = qt[cb + he * kW + w];
        const float td = qt[cb + (he + 1) * kW + w];
        const float gpv = sqrtf((pb - pa) * (pb - pa) + (pd - pc) * (pd - pc) + kEps);
        const float gtv = sqrtf((tb - ta) * (tb - ta) + (td - tc) * (td - tc) + kEps);
        grad_sum += fabsf(gpv - gtv);
    }
#pragma unroll
    for (int off = 16; off >= 1; off >>= 1) grad_sum += __shfl_xor(grad_sum, off, 32);

    // ---------------- block partial (deterministic, no atomics) ----------
    // loss = (rot_total + 0.05/4 * grad_total) / Npix
    __shared__ float smem[8];
    if (lane == 0) smem[wid] = rot_sum + (kWGrad * 0.25f) * grad_sum;
    __syncthreads();
    if (threadIdx.x == 0) {
        float s = 0.0f;
#pragma unroll
        for (int i = 0; i < 8; ++i) s += smem[i];
        partials[blockIdx.x] = s;
    }
}

__global__ __launch_bounds__(256)
void rotloss_finalize(const float* __restrict__ partials, int nPart,
                      float* __restrict__ out, float invN)
{
    __shared__ float smem[256];
    float s = 0.0f;
    for (int i = threadIdx.x; i < nPart; i += 256) s += partials[i];  // fixed order
    smem[threadIdx.x] = s;
    __syncthreads();
    for (int st = 128; st > 0; st >>= 1) {
        if (threadIdx.x < st) smem[threadIdx.x] += smem[threadIdx.x + st];
        __syncthreads();
    }
    if (threadIdx.x == 0) out[0] = smem[0] * invN;
}

extern "C" void kernel_launch(void* const* d_in, const int* in_sizes, int n_in,
                              void* d_out, int out_size, void* d_ws, size_t ws_size,
                              hipStream_t stream) {
    const float* q_pred   = (const float*)d_in[0];
    const float* q_target = (const float*)d_in[1];
    const float* syms     = (const float*)d_in[2];
    float* out      = (float*)d_out;
    float* partials = (float*)d_ws;

    const int npix   = in_sizes[0] / 4;   // 1,048,576
    const int blocks = npix / 256;        // 4096 (N divisible by 256 -> EXEC always full for WMMA)

    rotloss_partials<<<blocks, 256, 0, stream>>>(q_pred, q_target, syms, partials);
    rotloss_finalize<<<1, 256, 0, stream>>>(partials, blocks, out, 1.0f / (float)npix);
}